// SignatureModel_75144747810991
// MI455X (gfx1250) — compile-verified
//
#include <hip/hip_runtime.h>
#include <math.h>

// Problem constants (from reference)
#define TLEN   1024
#define NINC   1023           // increments per path
#define CDIM   12             // 1 + IN_CH + EXTRA
#define SIGC   1884           // 12 + 144 + 1728
#define NPATH  64             // B * GROUPS
#define NCHUNK 8              // Chen-parallel chunks per path
#define CHUNK_STEPS 128
#define KDIM   3768           // GROUPS * SIGC
#define OUTD   32
#define BATCH  32

typedef __attribute__((ext_vector_type(2))) float v2f;
typedef __attribute__((ext_vector_type(8))) float v8f;

// ---------------------------------------------------------------------------
// Phase 1: per-chunk depth-3 signature via sequential exp/Chen scan.
// One block per (path, chunk). Thread tid<144 owns (i,j): S2[i][j] and
// S3[i][j][0..11] in registers; S1 (12 floats) in LDS.
// Step (old S): S3 += E3 + S1(x)E2 + S2(x)E1 ; S2 += E2 + S1(x)E1 ; S1 += d
//   => S3[ijk] += d_k * (d_i*d_j/6 + S1_i*d_j/2 + S2_ij)
// aug bias cancels in increments; time increment is constant 1/1023.
// ---------------------------------------------------------------------------
__global__ __launch_bounds__(160) void sig_chunk_kernel(
    const float* __restrict__ x, const float* __restrict__ aug_w,
    float* __restrict__ csig)
{
  const int p = blockIdx.x >> 3;      // path 0..63
  const int c = blockIdx.x & 7;       // chunk 0..7
  const int b = p >> 1;
  const int g = p & 1;
  const int tid = threadIdx.x;
  const int i = tid / CDIM, j = tid % CDIM;

  __shared__ float s1[CDIM];

  float w[8][3];
  #pragma unroll
  for (int e = 0; e < 8; ++e)
    #pragma unroll
    for (int cc = 0; cc < 3; ++cc)
      w[e][cc] = aug_w[(g * 8 + e) * 3 + cc];

  if (tid < CDIM) s1[tid] = 0.0f;
  float s2 = 0.0f;
  float s3[CDIM];
  #pragma unroll
  for (int k = 0; k < CDIM; ++k) s3[k] = 0.0f;

  const int t0 = c * CHUNK_STEPS;
  const int t1 = (t0 + CHUNK_STEPS < NINC) ? (t0 + CHUNK_STEPS) : NINC;

  const float* xrow = x + (size_t)b * TLEN * 3;
  float xp0 = xrow[t0 * 3 + 0];
  float xp1 = xrow[t0 * 3 + 1];
  float xp2 = xrow[t0 * 3 + 2];
  const float dt = 1.0f / 1023.0f;
  __syncthreads();

  for (int t = t0; t < t1; ++t) {
    float x0 = xrow[(t + 1) * 3 + 0];
    float x1 = xrow[(t + 1) * 3 + 1];
    float x2 = xrow[(t + 1) * 3 + 2];
    float dx0 = x0 - xp0, dx1 = x1 - xp1, dx2 = x2 - xp2;
    xp0 = x0; xp1 = x1; xp2 = x2;

    float d[CDIM];
    d[0] = dt; d[1] = dx0; d[2] = dx1; d[3] = dx2;
    #pragma unroll
    for (int e = 0; e < 8; ++e)
      d[4 + e] = w[e][0] * dx0 + w[e][1] * dx1 + w[e][2] * dx2;

    if (tid < 144) {
      float s1i = s1[i];                 // old S1_i
      float di = d[i], dj = d[j];
      float a = di * dj * (1.0f / 6.0f) + s1i * dj * 0.5f + s2;
      #pragma unroll
      for (int k = 0; k < CDIM; ++k) s3[k] = fmaf(a, d[k], s3[k]);
      s2 = fmaf(s1i, dj, fmaf(di, dj * 0.5f, s2));
    }
    __syncthreads();                     // all old-S1 reads done
    if (tid < CDIM) s1[tid] += d[tid];
    __syncthreads();                     // new S1 visible for next step
  }

  // signatory layout: [S1(12) | S2 i*12+j (144) | S3 i*144+j*12+k (1728)]
  float* o = csig + ((size_t)p * NCHUNK + c) * SIGC;
  if (tid < CDIM) o[tid] = s1[tid];
  if (tid < 144) {
    o[CDIM + tid] = s2;
    #pragma unroll
    for (int k = 0; k < CDIM; ++k)
      o[CDIM + 144 + tid * CDIM + k] = s3[k];
  }
}

// ---------------------------------------------------------------------------
// Phase 2: Chen-combine the 8 chunk signatures of each path (left to right):
//   R1 = A1+B1 ; R2 = A2+B2+A1(x)B1 ; R3 = A3+B3+A1(x)B2+A2(x)B1
// Same (i,j) thread ownership as phase 1. Writes final sig row-block
// at sig[b][g*1884 + ...].
// ---------------------------------------------------------------------------
__global__ __launch_bounds__(160) void sig_combine_kernel(
    const float* __restrict__ csig, float* __restrict__ sig)
{
  const int p = blockIdx.x;
  const int b = p >> 1, g = p & 1;
  const int tid = threadIdx.x;
  const int i = tid / CDIM, j = tid % CDIM;
  __shared__ float a1[CDIM];

  const float* base = csig + (size_t)p * NCHUNK * SIGC;
  float r2 = 0.0f, r3[CDIM];
  #pragma unroll
  for (int k = 0; k < CDIM; ++k) r3[k] = 0.0f;

  if (tid < CDIM) a1[tid] = base[tid];
  if (tid < 144) {
    r2 = base[CDIM + tid];
    #pragma unroll
    for (int k = 0; k < CDIM; ++k) r3[k] = base[CDIM + 144 + tid * CDIM + k];
  }
  __syncthreads();

  for (int c = 1; c < NCHUNK; ++c) {
    const float* B = base + (size_t)c * SIGC;
    float b1[CDIM];
    #pragma unroll
    for (int k = 0; k < CDIM; ++k) b1[k] = B[k];

    if (tid < 144) {
      float a1i = a1[i];                               // old A1_i
      const float* B2row = B + CDIM + j * CDIM;        // B2[j][*]
      float b2ij = B[CDIM + i * CDIM + j];
      const float* B3 = B + CDIM + 144 + tid * CDIM;
      #pragma unroll
      for (int k = 0; k < CDIM; ++k)
        r3[k] += B3[k] + a1i * B2row[k] + r2 * b1[k];  // old A2 (=r2)
      r2 += b2ij + a1i * b1[j];
    }
    __syncthreads();
    if (tid < CDIM) a1[tid] += b1[tid];
    __syncthreads();
  }

  float* row = sig + (size_t)b * KDIM + (size_t)g * SIGC;
  if (tid < CDIM) row[tid] = a1[tid];
  if (tid < 144) {
    row[CDIM + tid] = r2;
    #pragma unroll
    for (int k = 0; k < CDIM; ++k)
      row[CDIM + 144 + tid * CDIM + k] = r3[k];
  }
}

// ---------------------------------------------------------------------------
// Phase 3: classifier head with native f32 WMMA.
// out(32x32) = sigmoid( sig(32x3768) @ lin_w^T(3768x32) + lin_b )
// 4 waves, one 16x16 tile each; K-loop in steps of 8 with two accumulators.
// ISA §7.12.2 32-bit layouts:
//   A 16x4 : lane<16 -> K={k,k+1}, lane>=16 -> K={k+2,k+3}, M=lane%16
//   B 4x16 : mirrored (VGPR pair = consecutive K, lane-half = +2K), N=lane%16
//   C/D    : c[r] -> row r + 8*(lane>=16), col lane%16
// Epilogue sigmoid uses v_exp_f32 + v_rcp_f32 (no IEEE div chain needed).
// ---------------------------------------------------------------------------
__global__ __launch_bounds__(128) void sig_linear_kernel(
    const float* __restrict__ sig, const float* __restrict__ lin_w,
    const float* __restrict__ lin_b, float* __restrict__ out)
{
  const int lane  = threadIdx.x & 31;
  const int wv    = threadIdx.x >> 5;     // 0..3 -> 2x2 tile grid
  const int mbase = (wv >> 1) * 16;
  const int nbase = (wv & 1) * 16;
  const int half  = lane >> 4;
  const int l     = lane & 15;
  const int m     = mbase + l;            // A row for this lane
  const int n     = nbase + l;            // B col for this lane

  const float* arow = sig   + (size_t)m * KDIM + 2 * half;
  const float* brow = lin_w + (size_t)n * KDIM + 2 * half;   // B[k][n] = lin_w[n][k]

  v8f c0 = {}; v8f c1 = {};
  for (int k = 0; k < KDIM; k += 8) {
    v2f a0 = *(const v2f*)(arow + k);
    v2f b0 = *(const v2f*)(brow + k);
    v2f a1 = *(const v2f*)(arow + k + 4);
    v2f b1 = *(const v2f*)(brow + k + 4);
    c0 = __builtin_amdgcn_wmma_f32_16x16x4_f32(false, a0, false, b0,
                                               (short)0, c0, false, false);
    c1 = __builtin_amdgcn_wmma_f32_16x16x4_f32(false, a1, false, b1,
                                               (short)0, c1, false, false);
  }

  float bias = lin_b[n];
  #pragma unroll
  for (int r = 0; r < 8; ++r) {
    int row = mbase + r + 8 * half;
    float v = c0[r] + c1[r] + bias;
    // sigmoid(v) = rcp(1 + exp(-v)); v_exp_f32 + v_rcp_f32, no div chain
    float e = __expf(-v);
    out[row * OUTD + (nbase + l)] = __builtin_amdgcn_rcpf(1.0f + e);
  }
}

// ---------------------------------------------------------------------------
extern "C" void kernel_launch(void* const* d_in, const int* in_sizes, int n_in,
                              void* d_out, int out_size, void* d_ws, size_t ws_size,
                              hipStream_t stream) {
  (void)in_sizes; (void)n_in; (void)out_size; (void)ws_size;

  const float* x     = (const float*)d_in[0];   // (32,1024,3)
  const float* aug_w = (const float*)d_in[1];   // (2,8,3)
  // d_in[2] = aug_b — unused: constant bias cancels in path increments.
  const float* lin_w = (const float*)d_in[3];   // (32,3768)
  const float* lin_b = (const float*)d_in[4];   // (32,)
  float* out = (float*)d_out;                   // (32,32)

  // Workspace: chunk signatures (64*8*1884 f32 = 3.86 MB) then final sig
  // matrix (32*3768 f32 = 0.48 MB). Both fully overwritten each call.
  float* csig = (float*)d_ws;
  float* sig  = csig + (size_t)NPATH * NCHUNK * SIGC;

  sig_chunk_kernel  <<<NPATH * NCHUNK, 160, 0, stream>>>(x, aug_w, csig);
  sig_combine_kernel<<<NPATH,          160, 0, stream>>>(csig, sig);
  sig_linear_kernel <<<1,              128, 0, stream>>>(sig, lin_w, lin_b, out);
}